// AdjacencyMatrixLayer_23510650978668
// MI455X (gfx1250) — compile-verified
//
#include <hip/hip_runtime.h>

// CDNA5 / gfx1250, wave32. Uses V_WMMA_F32_16X16X4_F32.
typedef float v2f __attribute__((ext_vector_type(2)));
typedef float v8f __attribute__((ext_vector_type(8)));

#define K_PTS 1024
#define TILE 16
#define WAVES 8
#define TPB 256            // 8 wave32 waves
#define TPW 8              // 64 column tiles / 8 waves

__global__ __launch_bounds__(TPB)
void adjacency_softmax_kernel(const float* __restrict__ coords,
                              const float* __restrict__ p2_ptr,
                              float* __restrict__ out)
{
    // Per-batch point cache: x, y, z, ||p||^2
    __shared__ float4 pts[K_PTS];                 // 16 KB
    __shared__ float  redsum[TILE][WAVES];        // per-row per-wave partial sums

    const int tid   = threadIdx.x;
    const int wave  = tid >> 5;
    const int lane  = tid & 31;
    const int lhalf = lane >> 4;                  // 0: lanes 0-15, 1: lanes 16-31
    const int l15   = lane & 15;

    const int wg    = blockIdx.x;
    const int batch = wg >> 6;                    // 64 row-strips per batch
    const int strip = wg & 63;
    const int row0  = strip * TILE;

    const float p2 = p2_ptr[0];

    // ---- Stage coords + norms for this batch into LDS (cooperative) ----
    const float* cb = coords + (size_t)batch * K_PTS * 3;
    for (int i = tid; i < K_PTS; i += TPB) {
        float x = cb[i * 3 + 0];
        float y = cb[i * 3 + 1];
        float z = cb[i * 3 + 2];
        pts[i] = make_float4(x, y, z, x * x + y * y + z * z);
    }
    __syncthreads();

    // ---- A fragment (16x4 f32): rows row0..row0+15 of this strip ----
    // Lanes 0-15 hold K=0,1 (x,y); lanes 16-31 hold K=2,3 (z, norm).
    const float4 pa = pts[row0 + l15];
    v2f afrag;
    afrag.x = lhalf ? pa.z : pa.x;
    afrag.y = lhalf ? pa.w : pa.y;

    float e[TPW][8];       // exp(A) tiles kept in VGPRs
    float partial[8];
#pragma unroll
    for (int r = 0; r < 8; ++r) partial[r] = 0.0f;

    // ---- Compute 8 contiguous 16x16 tiles per wave via WMMA ----
#pragma unroll
    for (int t = 0; t < TPW; ++t) {
        const int j0 = TILE * (wave * TPW + t);
        const float4 pb = pts[j0 + l15];

        // B fragment (4x16 f32): VGPR0 = K0/K2, VGPR1 = K1/K3 per lane half.
        // B col n = [-2x, -2y, -2z, 1]
        v2f bfrag;
        bfrag.x = lhalf ? (-2.0f * pb.z) : (-2.0f * pb.x);
        bfrag.y = lhalf ? 1.0f           : (-2.0f * pb.y);

        // C = ||x_n||^2 broadcast down the column (same for all 8 row-VGPRs)
        v8f cfrag;
#pragma unroll
        for (int r = 0; r < 8; ++r) cfrag[r] = pb.w;

        // D = A*B + C  ->  squared distances for this 16x16 tile
        v8f d = __builtin_amdgcn_wmma_f32_16x16x4_f32(
            /*neg_a=*/false, afrag, /*neg_b=*/false, bfrag,
            /*c_mod=*/(short)0, cfrag, /*reuse_a=*/false, /*reuse_b=*/false);

        // A = exp(-p2*dist) in (0,1]; softmax needs exp(A); no max-subtract needed.
#pragma unroll
        for (int r = 0; r < 8; ++r) {
            float aval = __expf(-p2 * d[r]);
            float ev   = __expf(aval);
            e[t][r]    = ev;
            partial[r] += ev;
        }
    }

    // ---- Row sums: butterfly within each 16-lane half (columns of the tile) ----
#pragma unroll
    for (int r = 0; r < 8; ++r) {
        float s = partial[r];
        s += __shfl_xor(s, 1, 32);
        s += __shfl_xor(s, 2, 32);
        s += __shfl_xor(s, 4, 32);
        s += __shfl_xor(s, 8, 32);
        partial[r] = s;               // all 16 lanes of the half hold the sum
    }
    if (l15 == 0) {
#pragma unroll
        for (int r = 0; r < 8; ++r)
            redsum[r + 8 * lhalf][wave] = partial[r];
    }
    __syncthreads();

    // ---- Combine across waves, form reciprocals for this lane's 8 rows ----
    float inv[8];
#pragma unroll
    for (int r = 0; r < 8; ++r) {
        const int row = r + 8 * lhalf;
        float s = 0.0f;
#pragma unroll
        for (int w = 0; w < WAVES; ++w) s += redsum[row][w];
        inv[r] = 1.0f / s;
    }

    // ---- Scale and stream out: write-once 512MB stream, mark non-temporal ----
    float* ob = out + (size_t)batch * K_PTS * K_PTS;
#pragma unroll
    for (int t = 0; t < TPW; ++t) {
        const int j = TILE * (wave * TPW + t) + l15;
#pragma unroll
        for (int r = 0; r < 8; ++r) {
            const int i = row0 + r + 8 * lhalf;
            __builtin_nontemporal_store(e[t][r] * inv[r],
                                        &ob[(size_t)i * K_PTS + j]);
        }
    }
}

extern "C" void kernel_launch(void* const* d_in, const int* in_sizes, int n_in,
                              void* d_out, int out_size, void* d_ws, size_t ws_size,
                              hipStream_t stream) {
    const float* coords = (const float*)d_in[0];
    const float* p2     = (const float*)d_in[1];
    float* out          = (float*)d_out;

    const int B = 128;
    const int strips = K_PTS / TILE;              // 64
    dim3 grid(B * strips);                        // 8192 workgroups
    dim3 block(TPB);
    adjacency_softmax_kernel<<<grid, block, 0, stream>>>(coords, p2, out);
}